// HeatmapGNN_32401233281101
// MI455X (gfx1250) — compile-verified
//
#include <hip/hip_runtime.h>
#include <hip/hip_bf16.h>

// ---------------- problem constants (match reference) ----------------
#define GNN_N      50000
#define GNN_E      800000
#define GNN_ET     (GNN_E + GNN_N)   // edges + self loops
#define GNN_IN     128
#define GNN_HID    64
#define GNN_HEADS  4
#define GNN_F      256               // HEADS*HID
#define GNN_OUT    64

typedef __attribute__((ext_vector_type(2))) float v2f;
typedef __attribute__((ext_vector_type(8))) float v8f;
typedef __attribute__((ext_vector_type(4))) unsigned u32x4;
typedef __attribute__((ext_vector_type(8))) int      i32x8;
typedef __attribute__((ext_vector_type(4))) int      i32x4;

// ---------------- helpers ----------------
// Monotone order-preserving float <-> unsigned encoding for atomicMax on floats
__device__ __forceinline__ unsigned fenc(float f) {
  unsigned u = __float_as_uint(f);
  return (u & 0x80000000u) ? ~u : (u | 0x80000000u);
}
__device__ __forceinline__ float fdec(unsigned u) {
  unsigned b = (u & 0x80000000u) ? (u ^ 0x80000000u) : ~u;
  return __uint_as_float(b);
}

__device__ __forceinline__ void edge_src_dst(const long long* __restrict__ ei, int e,
                                             int& s, int& d) {
  if (e < GNN_E) { s = (int)ei[e]; d = (int)ei[GNN_E + e]; }
  else           { s = e - GNN_E; d = s; }
}

// ---------------- generic zero ----------------
__global__ void gnn_zero_u32(unsigned* __restrict__ p, int n) {
  int i = blockIdx.x * blockDim.x + threadIdx.x;
  if (i < n) p[i] = 0u;
}

// ---------------- CSR build ----------------
__global__ void gnn_count_kernel(const long long* __restrict__ ei, int* __restrict__ counts) {
  int e = blockIdx.x * blockDim.x + threadIdx.x;
  if (e >= GNN_ET) return;
  int s, d; edge_src_dst(ei, e, s, d);
  atomicAdd(counts + d, 1);
}

// Single-workgroup chunked exclusive scan over N ints (N=50000 -> 49 chunks of 1024)
__global__ void gnn_scan_kernel(const int* __restrict__ counts, int* __restrict__ offs, int n) {
  __shared__ int buf[1024];
  __shared__ int carry;
  int tid = threadIdx.x;
  if (tid == 0) carry = 0;
  __syncthreads();
  for (int base = 0; base < n; base += 1024) {
    int i = base + tid;
    int v = (i < n) ? counts[i] : 0;
    buf[tid] = v;
    __syncthreads();
    for (int off = 1; off < 1024; off <<= 1) {
      int t = (tid >= off) ? buf[tid - off] : 0;
      __syncthreads();
      buf[tid] += t;
      __syncthreads();
    }
    int incl = buf[tid];
    int excl = incl - v;
    int cl = carry;
    if (i < n) offs[i] = cl + excl;
    __syncthreads();
    if (tid == 1023) carry = cl + incl;
    __syncthreads();
  }
  if (tid == 0) offs[n] = carry;
}

__global__ void gnn_fill_kernel(const long long* __restrict__ ei, const int* __restrict__ offs,
                                int* __restrict__ cursor, int* __restrict__ csr_src,
                                int* __restrict__ csr_eid) {
  int e = blockIdx.x * blockDim.x + threadIdx.x;
  if (e >= GNN_ET) return;
  int s, d; edge_src_dst(ei, e, s, d);
  int p = offs[d] + atomicAdd(cursor + d, 1);
  csr_src[p] = s;
  csr_eid[p] = e;
}

// ---------------- fp32 WMMA GEMM: Out[M,NOUT] = A[M,K] @ W[K,NOUT] (+bias) ----------------
// Block = 256 threads = 8 waves, all sharing one 16-column tile of W.
// The K x 16 tile of W is staged into LDS once per block by the Tensor Data Mover
// (tensor_load_to_lds, waited with s_wait_tensorcnt), then each wave runs a
// V_WMMA_F32_16X16X4_F32 K-loop on its own 16-row tile.
// A frag (16x4):  lane&15 = row, lane>>4 selects K pair {0,1}/{2,3} in 2 VGPRs
// B frag (4x16):  lane&15 = col, VGPR g + 2*(lane>>4) -> K row
// D (16x16):      VGPR g: lanes0-15 row=g, lanes16-31 row=g+8; col = lane&15
template <int K, int NOUT>
__global__ void gnn_gemm_wmma(const float* __restrict__ A, const float* __restrict__ W,
                              const float* __restrict__ bias, float* __restrict__ Out, int M) {
  __shared__ float Bs[K * 16];   // staged column tile of W (row-major K x 16)
  const int lane = threadIdx.x & 31;
  const int wave = threadIdx.x >> 5;
  const int c0 = blockIdx.x * 16;

  if (wave == 0) {
    // --- Tensor DMA descriptor (D#) for a K x 16 f32 tile of W, row stride NOUT ---
    unsigned lds_off = (unsigned)(size_t)(void*)Bs;                 // LDS byte offset
    unsigned long long ga = (unsigned long long)(size_t)(W + c0);   // tile start in global
    u32x4 g0;
    g0[0] = 1u;                                        // count=1, user-mode, no gather
    g0[1] = lds_off;                                   // lds_addr
    g0[2] = (unsigned)(ga & 0xffffffffull);            // global_addr[31:0]
    g0[3] = (unsigned)((ga >> 32) & 0x01ffffffull)     // global_addr[56:32]
            | (2u << 30);                              // type = 2 ("image")
    i32x8 g1;
    g1[0] = (int)(2u << 16);                           // data_size = 4B, no multicast
    g1[1] = (int)(16u << 16);                          // tensor_dim0[15:0] = 16 (bits 63:48)
    g1[2] = (int)(((unsigned)K & 0xffffu) << 16);      // tensor_dim1[15:0] = K (bits 95:80)
    g1[3] = (int)(16u << 16);                          // tile_dim0 = 16     (bits 127:112)
    g1[4] = (int)((unsigned)K & 0xffffu);              // tile_dim1 = K      (bits 143:128)
    g1[5] = (int)NOUT;                                 // tensor_dim0_stride[31:0] = NOUT
    g1[6] = 0;                                         // stride hi / dim1 stride
    g1[7] = 0;
    i32x4 gz4 = {0, 0, 0, 0};                          // groups 2/3 unused (2-D tensor)
    i32x8 gz8 = {0, 0, 0, 0, 0, 0, 0, 0};
    __builtin_amdgcn_tensor_load_to_lds(g0, g1, gz4, gz4, gz8, 0);
    __builtin_amdgcn_s_wait_tensorcnt(0);
  }
  __syncthreads();

  const int rowTile = blockIdx.y * 8 + wave;   // 16-row tile index
  if (rowTile * 16 < M) {
    const int r0 = rowTile * 16;
    const int m    = lane & 15;
    const int half = lane >> 4;
    v8f acc = {};
    const float* arow = A + (size_t)(r0 + m) * K + half * 2;
    const float* brow = Bs + half * 2 * 16 + m;
#pragma unroll 4
    for (int k = 0; k < K; k += 4) {
      v2f a, b;
      a.x = arow[k + 0];
      a.y = arow[k + 1];
      b.x = brow[k * 16];        // W[k + half*2 + 0][c0+m]
      b.y = brow[k * 16 + 16];   // W[k + half*2 + 1][c0+m]
      acc = __builtin_amdgcn_wmma_f32_16x16x4_f32(false, a, false, b, (short)0, acc,
                                                  false, false);
    }
    float bv = bias ? bias[c0 + m] : 0.0f;
#pragma unroll
    for (int g = 0; g < 8; ++g) {
      int row = r0 + g + half * 8;
      Out[(size_t)row * NOUT + (c0 + m)] = acc[g] + bv;
    }
  }
}

// ---------------- per-(node,head) attention coefficients ----------------
__global__ void gnn_alphas_kernel(const float* __restrict__ h, const float* __restrict__ a_src,
                                  const float* __restrict__ a_dst, float* __restrict__ as,
                                  float* __restrict__ ad) {
  int idx = blockIdx.x * blockDim.x + threadIdx.x;
  if (idx >= GNN_N * GNN_HEADS) return;
  int node = idx >> 2, head = idx & 3;
  const float4* hp = (const float4*)(h + (size_t)node * GNN_F + head * GNN_HID);
  const float4* sp = (const float4*)(a_src + head * GNN_HID);
  const float4* dp = (const float4*)(a_dst + head * GNN_HID);
  float s = 0.f, d = 0.f;
#pragma unroll 4
  for (int j = 0; j < GNN_HID / 4; ++j) {
    float4 hv = hp[j], sv = sp[j], dv = dp[j];
    s += hv.x * sv.x + hv.y * sv.y + hv.z * sv.z + hv.w * sv.w;
    d += hv.x * dv.x + hv.y * dv.y + hv.z * dv.z + hv.w * dv.w;
  }
  as[idx] = s;
  ad[idx] = d;
}

// ---------------- edge pass 1: leakyrelu logits + segment max (encoded atomicMax) ------
__global__ void gnn_edge_logits_kernel(const long long* __restrict__ ei,
                                       const float* __restrict__ as, const float* __restrict__ ad,
                                       float* __restrict__ ebuf, unsigned* __restrict__ menc) {
  int e = blockIdx.x * blockDim.x + threadIdx.x;
  if (e >= GNN_ET) return;
  int s, d; edge_src_dst(ei, e, s, d);
  float4 av = *(const float4*)(as + (size_t)s * 4);
  float4 dv = *(const float4*)(ad + (size_t)d * 4);
  float v[4] = {av.x + dv.x, av.y + dv.y, av.z + dv.z, av.w + dv.w};
#pragma unroll
  for (int hh = 0; hh < 4; ++hh) {
    float x = v[hh];
    x = x > 0.f ? x : 0.2f * x;   // LeakyReLU(0.2)
    v[hh] = x;
    atomicMax(menc + (size_t)d * 4 + hh, fenc(x));
  }
  *(float4*)(ebuf + (size_t)e * 4) = make_float4(v[0], v[1], v[2], v[3]);
}

// ---------------- edge pass 2: exp(e - max) + segment denom ----------------
__global__ void gnn_edge_exp_kernel(const long long* __restrict__ ei, float* __restrict__ ebuf,
                                    const unsigned* __restrict__ menc, float* __restrict__ denom) {
  int e = blockIdx.x * blockDim.x + threadIdx.x;
  if (e >= GNN_ET) return;
  int d = (e < GNN_E) ? (int)ei[GNN_E + e] : (e - GNN_E);
  float4 ev = *(const float4*)(ebuf + (size_t)e * 4);
  float v[4] = {ev.x, ev.y, ev.z, ev.w};
#pragma unroll
  for (int hh = 0; hh < 4; ++hh) {
    float mx = fdec(menc[(size_t)d * 4 + hh]);
    float x = __expf(v[hh] - mx);
    v[hh] = x;
    atomicAdd(denom + (size_t)d * 4 + hh, x);
  }
  *(float4*)(ebuf + (size_t)e * 4) = make_float4(v[0], v[1], v[2], v[3]);
}

// ---------------- CSR aggregation: one wave32 per destination node ----------------
// lane owns 8 contiguous channels (lane*8 .. lane*8+7), all in head = lane>>3.
// No scatter atomics: register accumulation, single coalesced row write, fused bias+ReLU.
// Next gathered row is prefetched (global_prefetch_b8) to hide L2 gather latency.
__global__ void gnn_aggregate_kernel(const float* __restrict__ hsrc, const float* __restrict__ ebuf,
                                     const float* __restrict__ denom, const int* __restrict__ offs,
                                     const int* __restrict__ csr_src, const int* __restrict__ csr_eid,
                                     const float* __restrict__ bias, float* __restrict__ out,
                                     int do_relu) {
  int wave = threadIdx.x >> 5;
  int lane = threadIdx.x & 31;
  int dstn = blockIdx.x * (blockDim.x >> 5) + wave;
  if (dstn >= GNN_N) return;
  int head = lane >> 3;
  float inv = 1.0f / denom[(size_t)dstn * 4 + head];
  int beg = offs[dstn], end = offs[dstn + 1];
  const int cbase = lane * 8;
  float acc[8] = {0.f, 0.f, 0.f, 0.f, 0.f, 0.f, 0.f, 0.f};
  for (int i = beg; i < end; ++i) {
    int srcn = csr_src[i];
    int eid  = csr_eid[i];
    if (i + 1 < end)
      __builtin_prefetch(hsrc + (size_t)csr_src[i + 1] * GNN_F + cbase, 0, 0);
    float w = ebuf[(size_t)eid * 4 + head] * inv;
    const float4* hp = (const float4*)(hsrc + (size_t)srcn * GNN_F + cbase);
    float4 h0 = hp[0], h1 = hp[1];
    acc[0] += w * h0.x; acc[1] += w * h0.y; acc[2] += w * h0.z; acc[3] += w * h0.w;
    acc[4] += w * h1.x; acc[5] += w * h1.y; acc[6] += w * h1.z; acc[7] += w * h1.w;
  }
#pragma unroll
  for (int j = 0; j < 8; ++j) {
    float v = acc[j] + bias[cbase + j];
    if (do_relu) v = fmaxf(v, 0.f);
    out[(size_t)dstn * GNN_F + cbase + j] = v;
  }
}

// ---------------- launch ----------------
extern "C" void kernel_launch(void* const* d_in, const int* in_sizes, int n_in,
                              void* d_out, int out_size, void* d_ws, size_t ws_size,
                              hipStream_t stream) {
  (void)in_sizes; (void)n_in; (void)out_size; (void)ws_size;
  const float*     x      = (const float*)d_in[0];
  const long long* ei     = (const long long*)d_in[1];   // int64 [2, E]
  const float*     W1     = (const float*)d_in[2];
  const float*     a_src1 = (const float*)d_in[3];
  const float*     a_dst1 = (const float*)d_in[4];
  const float*     b1     = (const float*)d_in[5];
  const float*     W2     = (const float*)d_in[6];
  const float*     a_src2 = (const float*)d_in[7];
  const float*     a_dst2 = (const float*)d_in[8];
  const float*     b2     = (const float*)d_in[9];
  const float*     Wf     = (const float*)d_in[10];
  const float*     bf     = (const float*)d_in[11];
  float*           out    = (float*)d_out;

  // workspace carve-up (256B aligned), ~127 MB total
  char* ws = (char*)d_ws;
  size_t off = 0;
  auto take = [&](size_t bytes) -> void* {
    void* p = ws + off;
    off += (bytes + 255) & ~(size_t)255;
    return p;
  };
  float*    hbuf    = (float*)   take((size_t)GNN_N * GNN_F * 4);
  float*    aggbuf  = (float*)   take((size_t)GNN_N * GNN_F * 4);
  float*    asb     = (float*)   take((size_t)GNN_N * 4 * 4);
  float*    adb     = (float*)   take((size_t)GNN_N * 4 * 4);
  unsigned* menc    = (unsigned*)take((size_t)GNN_N * 4 * 4);
  float*    denom   = (float*)   take((size_t)GNN_N * 4 * 4);
  float*    ebuf    = (float*)   take((size_t)GNN_ET * 4 * 4);
  int*      counts  = (int*)     take((size_t)GNN_N * 4);
  int*      offs    = (int*)     take((size_t)(GNN_N + 1) * 4);
  int*      cursor  = (int*)     take((size_t)GNN_N * 4);
  int*      csr_src = (int*)     take((size_t)GNN_ET * 4);
  int*      csr_eid = (int*)     take((size_t)GNN_ET * 4);

  const int T = 256;
  const int ebk = (GNN_ET + T - 1) / T;
  const int mTiles  = (GNN_N + 15) / 16;           // 3125
  const int gemmBy  = (mTiles + 7) / 8;            // 8 waves (row tiles) per block
  const dim3 gemm1Grid(GNN_F / 16, gemmBy);
  const dim3 gemmFGrid(GNN_OUT / 16, gemmBy);
  const int aggBlocks = (GNN_N + 7) / 8;

  // ---- CSR build (graph is identical for both layers) ----
  gnn_zero_u32<<<(GNN_N + T - 1) / T, T, 0, stream>>>((unsigned*)counts, GNN_N);
  gnn_count_kernel<<<ebk, T, 0, stream>>>(ei, counts);
  gnn_scan_kernel<<<1, 1024, 0, stream>>>(counts, offs, GNN_N);
  gnn_zero_u32<<<(GNN_N + T - 1) / T, T, 0, stream>>>((unsigned*)cursor, GNN_N);
  gnn_fill_kernel<<<ebk, T, 0, stream>>>(ei, offs, cursor, csr_src, csr_eid);

  // ---- layer 1 ----
  gnn_gemm_wmma<GNN_IN, GNN_F><<<gemm1Grid, T, 0, stream>>>(x, W1, nullptr, hbuf, GNN_N);
  gnn_alphas_kernel<<<(GNN_N * 4 + T - 1) / T, T, 0, stream>>>(hbuf, a_src1, a_dst1, asb, adb);
  gnn_zero_u32<<<(GNN_N * 4 + T - 1) / T, T, 0, stream>>>(menc, GNN_N * 4);
  gnn_zero_u32<<<(GNN_N * 4 + T - 1) / T, T, 0, stream>>>((unsigned*)denom, GNN_N * 4);
  gnn_edge_logits_kernel<<<ebk, T, 0, stream>>>(ei, asb, adb, ebuf, menc);
  gnn_edge_exp_kernel<<<ebk, T, 0, stream>>>(ei, ebuf, menc, denom);
  gnn_aggregate_kernel<<<aggBlocks, T, 0, stream>>>(hbuf, ebuf, denom, offs, csr_src, csr_eid,
                                                    b1, aggbuf, 1);

  // ---- layer 2 ----
  gnn_gemm_wmma<GNN_F, GNN_F><<<gemm1Grid, T, 0, stream>>>(aggbuf, W2, nullptr, hbuf, GNN_N);
  gnn_alphas_kernel<<<(GNN_N * 4 + T - 1) / T, T, 0, stream>>>(hbuf, a_src2, a_dst2, asb, adb);
  gnn_zero_u32<<<(GNN_N * 4 + T - 1) / T, T, 0, stream>>>(menc, GNN_N * 4);
  gnn_zero_u32<<<(GNN_N * 4 + T - 1) / T, T, 0, stream>>>((unsigned*)denom, GNN_N * 4);
  gnn_edge_logits_kernel<<<ebk, T, 0, stream>>>(ei, asb, adb, ebuf, menc);
  gnn_edge_exp_kernel<<<ebk, T, 0, stream>>>(ei, ebuf, menc, denom);
  gnn_aggregate_kernel<<<aggBlocks, T, 0, stream>>>(hbuf, ebuf, denom, offs, csr_src, csr_eid,
                                                    b2, aggbuf, 1);

  // ---- final projection: out = h2 @ Wf + bf ----
  gnn_gemm_wmma<GNN_F, GNN_OUT><<<gemmFGrid, T, 0, stream>>>(aggbuf, Wf, bf, out, GNN_N);
}